// WindowAttentionLayer_4741643894857
// MI455X (gfx1250) — compile-verified
//
#include <hip/hip_runtime.h>

typedef __attribute__((ext_vector_type(2))) float v2f;
typedef __attribute__((ext_vector_type(8))) float v8f;

#define TT    49          // tokens per window
#define TPAD  64          // padded to 4x16 tiles
#define CC    128         // channels
#define NH    4           // heads
#define HD    32          // head dim
#define NWIN  64          // number of masks
#define SX    132         // LDS row stride for x/q/k/v/out (bank-conflict pad)
#define SATT  68          // LDS row stride for attn tile

__device__ __forceinline__ v8f wmma_f32(v2f a, v2f b, v8f c) {
  // D = A(16x4) * B(4x16) + C(16x16), fp32
  return __builtin_amdgcn_wmma_f32_16x16x4_f32(false, a, false, b, (short)0, c,
                                               false, false);
}

__device__ __forceinline__ void store_cd_tile(v8f acc, float* dst, int stride,
                                              int mrow0, int col, int half,
                                              float addv, float mul, bool zpad) {
#pragma unroll
  for (int j = 0; j < 8; ++j) {
    int t = mrow0 + j + 8 * half;
    float val = (acc[j] + addv) * mul;
    if (zpad && t >= TT) val = 0.0f;
    dst[t * stride + col] = val;
  }
}

__global__ __launch_bounds__(256)
void swin_window_attn_kernel(const float* __restrict__ x,
                             const float* __restrict__ mask,
                             const float* __restrict__ qkv_w,
                             const float* __restrict__ qkv_b,
                             const float* __restrict__ proj_w,
                             const float* __restrict__ proj_b,
                             const float* __restrict__ bias_table,
                             float* __restrict__ out) {
  extern __shared__ float smem[];
  float* xbuf = smem;                          // 64*132 (later reused as out)
  float* qbuf = smem + (size_t)TPAD * SX;      // 64*132
  float* kbuf = smem + (size_t)2 * TPAD * SX;  // 64*132
  float* vbuf = smem + (size_t)3 * TPAD * SX;  // 64*132
  float* abuf = smem + (size_t)4 * TPAD * SX;  // 64*68
  float* bias_lds = abuf + (size_t)TPAD * SATT;  // 169*4 = 676

  const int b = blockIdx.x;
  const int tid = threadIdx.x;
  const int wid = tid >> 5;
  const int lane = tid & 31;
  const int half = lane >> 4;   // 0: K/M 0..; 1: K+2 / M+8 half
  const int r = lane & 15;
  const float scale = 0.17677669529663687f;    // 32^-0.5

  // ---- Phase A: stage x (zero padded) and bias table into LDS ----
  const float* xg = x + (size_t)b * TT * CC;
  for (int i = tid; i < TPAD * CC; i += 256) {
    int t = i >> 7, c = i & (CC - 1);
    xbuf[t * SX + c] = (t < TT) ? xg[t * CC + c] : 0.0f;
  }
  for (int i = tid; i < 169 * NH; i += 256) bias_lds[i] = bias_table[i];
  __syncthreads();

  // ---- Phase B: qkv = x @ qkv_w^T + qkv_b   (M=64, N=384, K=128) ----
  // wave `wid` owns n-tiles {wid, wid+8, wid+16}, all 4 m-tiles.
#pragma unroll
  for (int nti = 0; nti < 3; ++nti) {
    const int nbase = (wid + nti * 8) * 16;
    const int ncol = nbase + r;
    v8f acc0 = {}, acc1 = {}, acc2 = {}, acc3 = {};
#pragma unroll 4
    for (int kk = 0; kk < CC; kk += 4) {
      const int ko = kk + 2 * half;
      v2f bf = *(const v2f*)(qkv_w + (size_t)ncol * CC + ko);
      v2f a0 = *(const v2f*)(xbuf + (0 * 16 + r) * SX + ko);
      v2f a1 = *(const v2f*)(xbuf + (1 * 16 + r) * SX + ko);
      v2f a2 = *(const v2f*)(xbuf + (2 * 16 + r) * SX + ko);
      v2f a3 = *(const v2f*)(xbuf + (3 * 16 + r) * SX + ko);
      acc0 = wmma_f32(a0, bf, acc0);
      acc1 = wmma_f32(a1, bf, acc1);
      acc2 = wmma_f32(a2, bf, acc2);
      acc3 = wmma_f32(a3, bf, acc3);
    }
    const float bv = qkv_b[ncol];
    float* dst;
    int dc;
    float mul = 1.0f;
    if (nbase < CC) {               // q (scaled)
      dst = qbuf; dc = ncol; mul = scale;
    } else if (nbase < 2 * CC) {    // k
      dst = kbuf; dc = ncol - CC;
    } else {                        // v
      dst = vbuf; dc = ncol - 2 * CC;
    }
    store_cd_tile(acc0, dst, SX, 0,  dc, half, bv, mul, true);
    store_cd_tile(acc1, dst, SX, 16, dc, half, bv, mul, true);
    store_cd_tile(acc2, dst, SX, 32, dc, half, bv, mul, true);
    store_cd_tile(acc3, dst, SX, 48, dc, half, bv, mul, true);
  }
  __syncthreads();

  const float* mg = mask + (size_t)(b % NWIN) * TT * TT;

  // ---- Per-head attention ----
  for (int h = 0; h < NH; ++h) {
    const int hc = h * HD;
    // Phase C: attn = q_h @ k_h^T  (M=64, N=64, K=32); 16 tiles, 2 per wave
#pragma unroll
    for (int ti = 0; ti < 2; ++ti) {
      const int tile = wid * 2 + ti;
      const int mt = tile >> 2, nt = tile & 3;
      v8f acc = {};
#pragma unroll
      for (int kk = 0; kk < HD; kk += 4) {
        const int ko = hc + kk + 2 * half;
        v2f af = *(const v2f*)(qbuf + (mt * 16 + r) * SX + ko);
        v2f bf = *(const v2f*)(kbuf + (nt * 16 + r) * SX + ko);
        acc = wmma_f32(af, bf, acc);
      }
      const int s = nt * 16 + r;
      const int sh = s / 7, sw = s % 7;
#pragma unroll
      for (int j = 0; j < 8; ++j) {
        const int t = mt * 16 + j + 8 * half;
        float val = 0.0f;
        if (t < TT && s < TT) {
          const int th = t / 7, tw = t % 7;
          const int idx = (th - sh + 6) * 13 + (tw - sw + 6);
          val = acc[j] + bias_lds[idx * NH + h] + mg[t * TT + s];
        }
        abuf[t * SATT + s] = val;
      }
    }
    __syncthreads();
    // Phase D: out_h = attn @ v_h  (M=64, N=32, K=64); 8 tiles, 1 per wave
    {
      const int mt = wid >> 1, nt = wid & 1;
      const int dcol = hc + nt * 16 + r;
      v8f acc = {};
#pragma unroll 4
      for (int kk = 0; kk < TPAD; kk += 4) {
        const int krow = kk + 2 * half;
        v2f af = *(const v2f*)(abuf + (mt * 16 + r) * SATT + krow);
        v2f bf;
        bf.x = vbuf[krow * SX + dcol];
        bf.y = vbuf[(krow + 1) * SX + dcol];
        acc = wmma_f32(af, bf, acc);
      }
      // out columns are disjoint per head: write straight into xbuf (reused)
      store_cd_tile(acc, xbuf, SX, mt * 16, dcol, half, 0.0f, 1.0f, false);
    }
    __syncthreads();
  }

  // ---- Phase E: final = out @ proj_w^T + proj_b  (M=64, N=128, K=128) ----
  {
    const int ncol = wid * 16 + r;     // wave `wid` owns n-tile wid
    v8f acc0 = {}, acc1 = {}, acc2 = {}, acc3 = {};
#pragma unroll 4
    for (int kk = 0; kk < CC; kk += 4) {
      const int ko = kk + 2 * half;
      v2f bf = *(const v2f*)(proj_w + (size_t)ncol * CC + ko);
      v2f a0 = *(const v2f*)(xbuf + (0 * 16 + r) * SX + ko);
      v2f a1 = *(const v2f*)(xbuf + (1 * 16 + r) * SX + ko);
      v2f a2 = *(const v2f*)(xbuf + (2 * 16 + r) * SX + ko);
      v2f a3 = *(const v2f*)(xbuf + (3 * 16 + r) * SX + ko);
      acc0 = wmma_f32(a0, bf, acc0);
      acc1 = wmma_f32(a1, bf, acc1);
      acc2 = wmma_f32(a2, bf, acc2);
      acc3 = wmma_f32(a3, bf, acc3);
    }
    const float pb = proj_b[ncol];
    float* og = out + (size_t)b * TT * CC;
#pragma unroll
    for (int mt = 0; mt < 4; ++mt) {
      v8f acc = (mt == 0) ? acc0 : (mt == 1) ? acc1 : (mt == 2) ? acc2 : acc3;
#pragma unroll
      for (int j = 0; j < 8; ++j) {
        const int t = mt * 16 + j + 8 * half;
        if (t < TT) og[t * CC + ncol] = acc[j] + pb;
      }
    }
  }
}

extern "C" void kernel_launch(void* const* d_in, const int* in_sizes, int n_in,
                              void* d_out, int out_size, void* d_ws,
                              size_t ws_size, hipStream_t stream) {
  const float* x          = (const float*)d_in[0];
  const float* mask       = (const float*)d_in[1];
  const float* qkv_w      = (const float*)d_in[2];
  const float* qkv_b      = (const float*)d_in[3];
  const float* proj_w     = (const float*)d_in[4];
  const float* proj_b     = (const float*)d_in[5];
  const float* bias_table = (const float*)d_in[6];
  float* out = (float*)d_out;

  const int B = in_sizes[0] / (TT * CC);   // 8192 windows
  const size_t shmem =
      (size_t)(4 * TPAD * SX + TPAD * SATT + 169 * NH) * sizeof(float); // ~152KB

  hipFuncSetAttribute((const void*)swin_window_attn_kernel,
                      hipFuncAttributeMaxDynamicSharedMemorySize, (int)shmem);
  swin_window_attn_kernel<<<B, 256, shmem, stream>>>(
      x, mask, qkv_w, qkv_b, proj_w, proj_b, bias_table, out);
}